// MatrixModel_12884901888386
// MI455X (gfx1250) — compile-verified
//
#include <hip/hip_runtime.h>

// ---------------------------------------------------------------------------
// Masked-abs row softmax, [8191 x 16382] fp32.
//   W[i,j] = (8192+i > j) ? |w[i,j]| : 0 ;  out = softmax(W, axis=1)
// Bandwidth-bound: ~537MB read (-25% masked-tail loads skipped) + 537MB write.
// One block per row, 512 threads (16 wave32), row held in registers.
// Loads are split: k=0..7 unconditional (always kept -> 8 back-to-back b64
// loads, no waits), k=8..15 load-only branches (no consumer in the branch,
// so no forced s_wait_loadcnt; fully-masked waves skip the traffic).
// ---------------------------------------------------------------------------

typedef __attribute__((ext_vector_type(2))) float f32x2;

constexpr int kRows     = 8191;
constexpr int kCols     = 16382;
constexpr int kNumTerms = 8192;
constexpr int kBlock    = 512;
constexpr int kWaves    = kBlock / 32;            // 16
constexpr int kF2PerRow = kCols / 2;              // 8191 float2 per row
constexpr int kIters    = (kF2PerRow + kBlock - 1) / kBlock;  // 16
constexpr int kSafe     = 8;  // iters 0..7: columns < 8192 <= kept, always loaded

__device__ __forceinline__ float wave_reduce_max(float x) {
#pragma unroll
  for (int off = 16; off > 0; off >>= 1)
    x = fmaxf(x, __shfl_xor(x, off, 32));
  return x;
}

__device__ __forceinline__ float wave_reduce_sum(float x) {
#pragma unroll
  for (int off = 16; off > 0; off >>= 1)
    x += __shfl_xor(x, off, 32);
  return x;
}

__global__ __launch_bounds__(kBlock) void masked_abs_softmax(
    const float* __restrict__ W, float* __restrict__ O) {
  __shared__ float s_part[kWaves];
  __shared__ float s_bcast[2];

  const int row  = blockIdx.x;
  const int tid  = threadIdx.x;
  const int lane = tid & 31;
  const int wave = tid >> 5;
  const int kept = kNumTerms + row;  // keep column j iff j < kept

  const f32x2* __restrict__ Wrow =
      reinterpret_cast<const f32x2*>(W + (long long)row * kCols);
  f32x2* __restrict__ Orow =
      reinterpret_cast<f32x2*>(O + (long long)row * kCols);

  f32x2 r[kIters];

  // ---- Load phase A: always-kept region, branch-free, pipelined ----
#pragma unroll
  for (int k = 0; k < kSafe; ++k) {
    r[k] = __builtin_nontemporal_load(&Wrow[tid + k * kBlock]);
  }

  // ---- Load phase B: maskable tail; branch bodies contain ONLY the load ----
  // Pre-zeroed destinations mean skipped lanes contribute 0 (the masked
  // value) with no extra select later. Fully-masked waves skip the load.
#pragma unroll
  for (int k = kSafe; k < kIters; ++k) {
    r[k].x = 0.0f;
    r[k].y = 0.0f;
    const int idx = tid + k * kBlock;
    if (idx * 2 < kept) {  // implies idx < kF2PerRow
      r[k] = __builtin_nontemporal_load(&Wrow[idx]);
    }
  }

  // ---- Mask + abs + local max (branch-free) ----
  // All values >= 0 (abs or masked 0), so 0 is a valid max identity.
  float lmax = 0.0f;
#pragma unroll
  for (int k = 0; k < kSafe; ++k) {
    const float a = fabsf(r[k].x);
    const float b = fabsf(r[k].y);
    r[k].x = a;
    r[k].y = b;
    lmax = fmaxf(lmax, fmaxf(a, b));
  }
#pragma unroll
  for (int k = kSafe; k < kIters; ++k) {
    const int j1 = (tid + k * kBlock) * 2 + 1;
    const float a = fabsf(r[k].x);                         // unloaded lanes: |0| = 0
    const float b = (j1 < kept) ? fabsf(r[k].y) : 0.0f;    // odd-boundary element
    r[k].x = a;
    r[k].y = b;
    lmax = fmaxf(lmax, fmaxf(a, b));
  }

  // ---- Block max reduction (wave32 shuffle + tiny LDS) ----
  float wm = wave_reduce_max(lmax);
  if (lane == 0) s_part[wave] = wm;
  __syncthreads();
  if (wave == 0) {
    float y = (lane < kWaves) ? s_part[lane] : 0.0f;
    y = wave_reduce_max(y);
    if (lane == 0) s_bcast[0] = y;
  }
  __syncthreads();
  const float m = s_bcast[0];

  // ---- exp (v_exp_f32 via __expf) + local sum ----
  // Only the phantom slot (idx == 8191: thread 511, k == 15) is excluded.
  float lsum = 0.0f;
#pragma unroll
  for (int k = 0; k < kIters; ++k) {
    const float e0 = __expf(r[k].x - m);
    const float e1 = __expf(r[k].y - m);
    r[k].x = e0;
    r[k].y = e1;
    if (k < kIters - 1) {
      lsum += e0 + e1;
    } else {
      const int idx = tid + k * kBlock;
      lsum += (idx < kF2PerRow) ? (e0 + e1) : 0.0f;
    }
  }

  // ---- Block sum reduction ----
  float ws = wave_reduce_sum(lsum);
  __syncthreads();  // s_part reuse: all waves are past the max-phase read
  if (lane == 0) s_part[wave] = ws;
  __syncthreads();
  if (wave == 0) {
    float y = (lane < kWaves) ? s_part[lane] : 0.0f;
    y = wave_reduce_sum(y);
    if (lane == 0) s_bcast[1] = y;
  }
  __syncthreads();
  const float inv = 1.0f / s_bcast[1];

  // ---- Normalize + NT store; only k == 15 needs the range guard ----
#pragma unroll
  for (int k = 0; k < kIters - 1; ++k) {
    f32x2 o;
    o.x = r[k].x * inv;
    o.y = r[k].y * inv;
    __builtin_nontemporal_store(o, &Orow[tid + k * kBlock]);
  }
  {
    const int idx = tid + (kIters - 1) * kBlock;
    if (idx < kF2PerRow) {
      f32x2 o;
      o.x = r[kIters - 1].x * inv;
      o.y = r[kIters - 1].y * inv;
      __builtin_nontemporal_store(o, &Orow[idx]);
    }
  }
}

extern "C" void kernel_launch(void* const* d_in, const int* in_sizes, int n_in,
                              void* d_out, int out_size, void* d_ws,
                              size_t ws_size, hipStream_t stream) {
  (void)in_sizes; (void)n_in; (void)out_size; (void)d_ws; (void)ws_size;
  const float* W = reinterpret_cast<const float*>(d_in[0]);
  float* O = reinterpret_cast<float*>(d_out);
  masked_abs_softmax<<<dim3(kRows), dim3(kBlock), 0, stream>>>(W, O);
}